// SmoothedMetricLoss_31198642438473
// MI455X (gfx1250) — compile-verified
//
#include <hip/hip_runtime.h>
#include <math.h>

typedef float v2f __attribute__((ext_vector_type(2)));
typedef float v8f __attribute__((ext_vector_type(8)));

#define P_HALF 2048
#define DFEAT  256
#define TILE   32
#define NTILE  (P_HALF / TILE)   // 64 tiles per dimension

// ---------------------------------------------------------------------------
// Zero the row/col accumulators (4096 floats) each launch.
// ---------------------------------------------------------------------------
__global__ __launch_bounds__(256) void zero_ws_kernel(float* __restrict__ ws) {
    int i = blockIdx.x * 256 + threadIdx.x;
    if (i < 2 * P_HALF) ws[i] = 0.0f;
}

// ---------------------------------------------------------------------------
// Fused E = X_even * X_odd^T  (2048x2048x256, fp32 WMMA 16x16x4) with
// exp(1 + E/128) masked row/col reductions and diagonal capture.
// Each wave computes a 32x32 tile (2x2 accumulators of 16x16).
// ---------------------------------------------------------------------------
__global__ __launch_bounds__(256) void gram_exp_kernel(const float* __restrict__ in,
                                                       float* __restrict__ rowSum,
                                                       float* __restrict__ colSum,
                                                       float* __restrict__ dpos) {
    const int lane  = threadIdx.x & 31;
    const int wave  = threadIdx.x >> 5;
    const int gw    = blockIdx.x * 8 + wave;          // 0..4095
    const int tR    = gw >> 6;                        // /NTILE
    const int tC    = gw & (NTILE - 1);
    const int R0    = tR * TILE;
    const int C0    = tC * TILE;
    const int l15   = lane & 15;
    const int lhi   = lane >> 4;                      // 0 or 1
    const int half8 = (lane < 16) ? 0 : 8;

    // A fragment rows: E-rows R0..R0+31  -> input rows 2*(R0+m)
    // B fragment rows: E-cols C0..C0+31  -> input rows 2*(C0+n)+1
    const float* a0p = in + (size_t)(2 * (R0 + l15)) * DFEAT + 2 * lhi;
    const float* a1p = a0p + (size_t)32 * DFEAT;      // +16 E-rows = +32 input rows
    const float* b0p = in + (size_t)(2 * (C0 + l15) + 1) * DFEAT + 2 * lhi;
    const float* b1p = b0p + (size_t)32 * DFEAT;

    v8f c00 = {}; v8f c01 = {}; v8f c10 = {}; v8f c11 = {};

    #pragma unroll 4
    for (int k = 0; k < DFEAT; k += 4) {
        v2f a0 = *(const v2f*)(a0p + k);
        v2f a1 = *(const v2f*)(a1p + k);
        v2f b0 = *(const v2f*)(b0p + k);
        v2f b1 = *(const v2f*)(b1p + k);
        // D = A*B + C   (args: neg_a, A, neg_b, B, c_mod, C, reuse_a, reuse_b)
        c00 = __builtin_amdgcn_wmma_f32_16x16x4_f32(false, a0, false, b0, (short)0, c00, false, false);
        c01 = __builtin_amdgcn_wmma_f32_16x16x4_f32(false, a0, false, b1, (short)0, c01, false, false);
        c10 = __builtin_amdgcn_wmma_f32_16x16x4_f32(false, a1, false, b0, (short)0, c10, false, false);
        c11 = __builtin_amdgcn_wmma_f32_16x16x4_f32(false, a1, false, b1, (short)0, c11, false, false);
    }

    // ---- Epilogue: t = exp(1 + e/128); capture diagonal; masked reductions.
    // C/D layout: lane<16 -> (M = r,   N = l15); lane>=16 -> (M = r+8, N = l15)
    float tv[2][2][8];
    #pragma unroll
    for (int i = 0; i < 2; ++i) {
        #pragma unroll
        for (int j = 0; j < 2; ++j) {
            #pragma unroll
            for (int r = 0; r < 8; ++r) {
                float e = (i == 0) ? ((j == 0) ? c00[r] : c01[r])
                                   : ((j == 0) ? c10[r] : c11[r]);
                int gM = R0 + 16 * i + r + half8;
                int gN = C0 + 16 * j + l15;
                if (gM == gN) dpos[gM] = e * 0.0078125f;   // D[2p,2p+1]
                tv[i][j][r] = __expf(1.0f + e * 0.0078125f);
            }
        }
    }

    // Row sums: exclude c == 2r+1. Reduce across the 16-lane group; lanes 0
    // and 16 hold rows M=r and M=r+8 respectively.
    #pragma unroll
    for (int i = 0; i < 2; ++i) {
        #pragma unroll
        for (int r = 0; r < 8; ++r) {
            int gM = R0 + 16 * i + r + half8;
            float v = 0.0f;
            #pragma unroll
            for (int j = 0; j < 2; ++j) {
                int gN = C0 + 16 * j + l15;
                if (gN != 2 * gM + 1) v += tv[i][j][r];
            }
            #pragma unroll
            for (int off = 8; off; off >>= 1) v += __shfl_xor(v, off, 16);
            if (l15 == 0) atomicAdd(&rowSum[gM], v);
        }
    }

    // Col sums: exclude r == 2c. Sum 16 in-lane elements, then fold the two
    // half-wave row groups together.
    #pragma unroll
    for (int j = 0; j < 2; ++j) {
        int gN = C0 + 16 * j + l15;
        float v = 0.0f;
        #pragma unroll
        for (int i = 0; i < 2; ++i) {
            #pragma unroll
            for (int r = 0; r < 8; ++r) {
                int gM = R0 + 16 * i + r + half8;
                if (gM != 2 * gN) v += tv[i][j][r];
            }
        }
        v += __shfl_xor(v, 16, 32);
        if (lane < 16) atomicAdd(&colSum[gN], v);
    }
}

// ---------------------------------------------------------------------------
// J[p] = log(1e-8 + rowSum + colSum) - dpos; loss = sum(max(J,0)^2)/4096
// ---------------------------------------------------------------------------
__global__ __launch_bounds__(256) void finalize_kernel(const float* __restrict__ rowSum,
                                                       const float* __restrict__ colSum,
                                                       const float* __restrict__ dpos,
                                                       float* __restrict__ out) {
    __shared__ float red[256];
    float acc = 0.0f;
    for (int p = threadIdx.x; p < P_HALF; p += 256) {
        float J = logf(1e-8f + rowSum[p] + colSum[p]) - dpos[p];
        float m = fmaxf(J, 0.0f);
        acc += m * m;
    }
    red[threadIdx.x] = acc;
    __syncthreads();
    #pragma unroll
    for (int s = 128; s; s >>= 1) {
        if (threadIdx.x < s) red[threadIdx.x] += red[threadIdx.x + s];
        __syncthreads();
    }
    if (threadIdx.x == 0) out[0] = red[0] * (1.0f / 4096.0f);
}

extern "C" void kernel_launch(void* const* d_in, const int* in_sizes, int n_in,
                              void* d_out, int out_size, void* d_ws, size_t ws_size,
                              hipStream_t stream) {
    const float* in = (const float*)d_in[0];
    float* ws     = (float*)d_ws;
    float* rowSum = ws;                 // 2048 floats
    float* colSum = ws + P_HALF;        // 2048 floats
    float* dpos   = ws + 2 * P_HALF;    // 2048 floats (fully overwritten)

    zero_ws_kernel<<<(2 * P_HALF + 255) / 256, 256, 0, stream>>>(ws);
    gram_exp_kernel<<<(NTILE * NTILE) / 8, 256, 0, stream>>>(in, rowSum, colSum, dpos);
    finalize_kernel<<<1, 256, 0, stream>>>(rowSum, colSum, dpos, (float*)d_out);
}